// SelectiveSSM_47390669144424
// MI455X (gfx1250) — compile-verified
//
#include <hip/hip_runtime.h>
#include <hip/hip_bf16.h>
#include <math.h>

// Problem constants (match reference)
#define BATCH   2
#define SEQLEN  2048
#define DMODEL  768
#define DINNER  1536
#define DSTATE  16
#define DTRANK  48
#define DCONV   4
#define NTOK    (BATCH * SEQLEN)        // 4096 tokens
#define XDBL_W  (DTRANK + 2 * DSTATE)   // 80

typedef __attribute__((ext_vector_type(2))) float v2f;
typedef __attribute__((ext_vector_type(8))) float v8f;

__device__ __forceinline__ float silu_(float v)     { return v / (1.f + __expf(-v)); }
__device__ __forceinline__ float softplus_(float v) { return v > 20.f ? v : log1pf(__expf(v)); }

__device__ __forceinline__ v8f wmma_f32(v2f a, v2f b, v8f c) {
    return __builtin_amdgcn_wmma_f32_16x16x4_f32(
        /*neg_a=*/false, a, /*neg_b=*/false, b,
        /*c_mod=*/(short)0, c, /*reuse_a=*/false, /*reuse_b=*/false);
}

// ---------------------------------------------------------------------------
// LayerNorm: one wave per row of 768. 24 elements per lane (wave32).
// ---------------------------------------------------------------------------
__global__ __launch_bounds__(256) void ln_kernel(const float* __restrict__ x,
                                                 const float* __restrict__ w,
                                                 const float* __restrict__ b,
                                                 float* __restrict__ xn) {
    int row  = blockIdx.x * 8 + (threadIdx.x >> 5);
    int lane = threadIdx.x & 31;
    const float* xr = x + (size_t)row * DMODEL;
    float vals[24];
    float s = 0.f, s2 = 0.f;
#pragma unroll
    for (int i = 0; i < 24; ++i) {
        float v = xr[lane + i * 32];
        vals[i] = v; s += v; s2 += v * v;
    }
#pragma unroll
    for (int off = 16; off > 0; off >>= 1) {
        s  += __shfl_xor(s,  off);
        s2 += __shfl_xor(s2, off);
    }
    float mu   = s * (1.f / DMODEL);
    float var  = s2 * (1.f / DMODEL) - mu * mu;
    float rstd = rsqrtf(var + 1e-5f);
    float* o = xn + (size_t)row * DMODEL;
#pragma unroll
    for (int i = 0; i < 24; ++i) {
        int c = lane + i * 32;
        o[c] = (vals[i] - mu) * rstd * w[c] + b[c];
    }
}

// ---------------------------------------------------------------------------
// Register-blocked fp32 WMMA GEMM: C[M,N] = A[M,K] * W[N,K]^T (+ epilogue).
// One wave computes a 32x64 output tile: 2 A-frags x 4 B-frags -> 8 WMMAs
// per K-step from 6 b64 loads (vs 2 loads/WMMA unblocked). 8 v8f
// accumulators interleaved so consecutive WMMAs never share an accumulator.
// 8 waves/block; grids sized so every wave owns a valid tile (EXEC all-ones).
// fp32 frag layout (ISA 7.12.2): lane l holds row (l&15), K pair (l>>4)*2.
// EPI: 0 = none, 1 = softplus(v + bias[col]), 2 = v + resid[row,col]
// ---------------------------------------------------------------------------
template <int EPI>
__global__ __launch_bounds__(256) void gemm_wmma_32x64(
    const float* __restrict__ A, int lda,
    const float* __restrict__ W, int ldw,
    float* __restrict__ C, int ldc,
    int tilesN, int K,
    const float* __restrict__ bias,
    const float* __restrict__ resid, int ldr) {
    int wave = threadIdx.x >> 5;
    int lane = threadIdx.x & 31;
    int tile = blockIdx.x * 8 + wave;
    int tm   = (tile / tilesN) * 32;
    int tn   = (tile % tilesN) * 64;
    int half = lane >> 4;    // 0: K pair {0,1}, 1: K pair {2,3}
    int l16  = lane & 15;

    const float* Ap0 = A + (size_t)(tm + l16) * lda + (half << 1);
    const float* Ap1 = Ap0 + (size_t)16 * lda;
    const float* Wp0 = W + (size_t)(tn + l16) * ldw + (half << 1);
    const float* Wp1 = Wp0 + (size_t)16 * ldw;
    const float* Wp2 = Wp0 + (size_t)32 * ldw;
    const float* Wp3 = Wp0 + (size_t)48 * ldw;

    v8f acc[2][4] = {};
#pragma unroll 4
    for (int k = 0; k < K; k += 4) {
        v2f a0 = *(const v2f*)(Ap0 + k);
        v2f a1 = *(const v2f*)(Ap1 + k);
        v2f b0 = *(const v2f*)(Wp0 + k);
        v2f b1 = *(const v2f*)(Wp1 + k);
        v2f b2 = *(const v2f*)(Wp2 + k);
        v2f b3 = *(const v2f*)(Wp3 + k);
        acc[0][0] = wmma_f32(a0, b0, acc[0][0]);
        acc[1][0] = wmma_f32(a1, b0, acc[1][0]);
        acc[0][1] = wmma_f32(a0, b1, acc[0][1]);
        acc[1][1] = wmma_f32(a1, b1, acc[1][1]);
        acc[0][2] = wmma_f32(a0, b2, acc[0][2]);
        acc[1][2] = wmma_f32(a1, b2, acc[1][2]);
        acc[0][3] = wmma_f32(a0, b3, acc[0][3]);
        acc[1][3] = wmma_f32(a1, b3, acc[1][3]);
    }

#pragma unroll
    for (int mi = 0; mi < 2; ++mi) {
#pragma unroll
        for (int ni = 0; ni < 4; ++ni) {
            int col = tn + ni * 16 + l16;
#pragma unroll
            for (int v = 0; v < 8; ++v) {
                int row = tm + mi * 16 + (half << 3) + v;
                float val = acc[mi][ni][v];
                if (EPI == 1) val = softplus_(val + bias[col]);
                if (EPI == 2) val += resid[(size_t)row * ldr + col];
                C[(size_t)row * ldc + col] = val;
            }
        }
    }
}

// ---------------------------------------------------------------------------
// Narrow-N fp32 WMMA GEMM (16x16 tile per wave) for x_proj (N = 80 = 5*16).
// ---------------------------------------------------------------------------
__global__ __launch_bounds__(256) void gemm_wmma_16x16(
    const float* __restrict__ A, int lda,
    const float* __restrict__ W, int ldw,
    float* __restrict__ C, int ldc,
    int tilesN, int K) {
    int wave = threadIdx.x >> 5;
    int lane = threadIdx.x & 31;
    int tile = blockIdx.x * 8 + wave;
    int tm   = (tile / tilesN) << 4;
    int tn   = (tile % tilesN) << 4;
    int half = lane >> 4;
    int l16  = lane & 15;

    const float* Ap = A + (size_t)(tm + l16) * lda + (half << 1);
    const float* Wp = W + (size_t)(tn + l16) * ldw + (half << 1);

    v8f acc = {};
#pragma unroll 8
    for (int k = 0; k < K; k += 4) {
        v2f af = *(const v2f*)(Ap + k);
        v2f bf = *(const v2f*)(Wp + k);
        acc = wmma_f32(af, bf, acc);
    }
    int col = tn + l16;
#pragma unroll
    for (int v = 0; v < 8; ++v) {
        int row = tm + (half << 3) + v;
        C[(size_t)row * ldc + col] = acc[v];
    }
}

// ---------------------------------------------------------------------------
// Depthwise causal conv (width 4) + bias + SiLU. One thread per (b,l,d).
// x_in = columns [0,1536) of xz.
// ---------------------------------------------------------------------------
__global__ __launch_bounds__(256) void conv_silu_kernel(const float* __restrict__ xz,
                                                        const float* __restrict__ cw,
                                                        const float* __restrict__ cb,
                                                        float* __restrict__ xc) {
    int idx = blockIdx.x * blockDim.x + threadIdx.x;   // over NTOK*DINNER
    int d   = idx % DINNER;
    int bl  = idx / DINNER;
    int l   = bl % SEQLEN;
    int b   = bl / SEQLEN;
    float acc = cb[d];
#pragma unroll
    for (int j = 0; j < DCONV; ++j) {
        int ll = l - (DCONV - 1) + j;
        if (ll >= 0)
            acc += cw[d * DCONV + j] *
                   xz[((size_t)b * SEQLEN + ll) * (2 * DINNER) + d];
    }
    xc[idx] = silu_(acc);
}

// ---------------------------------------------------------------------------
// Selective scan. One lane per (b, d): 16-state linear recurrence over L,
// fully register resident (16 independent FMA chains per lane for ILP).
// Fuses the D-skip term and the SiLU(z) gate. Writes y in place over xc
// (each element is read before it is written, per-thread).
// ---------------------------------------------------------------------------
__global__ __launch_bounds__(256) void scan_kernel(const float* __restrict__ delta,
                                                   const float* __restrict__ xdbl,
                                                   const float* __restrict__ xz,
                                                   const float* __restrict__ xc,
                                                   const float* __restrict__ A_log,
                                                   const float* __restrict__ Dp,
                                                   float* __restrict__ y) {
    int gid = blockIdx.x * blockDim.x + threadIdx.x;   // BATCH*DINNER threads
    int d = gid % DINNER;
    int b = gid / DINNER;

    float a[DSTATE];
#pragma unroll
    for (int n = 0; n < DSTATE; ++n) a[n] = -__expf(A_log[d * DSTATE + n]);

    float h[DSTATE];
#pragma unroll
    for (int n = 0; n < DSTATE; ++n) h[n] = 0.f;

    float dcoef = Dp[d];

    for (int l = 0; l < SEQLEN; ++l) {
        size_t row = (size_t)b * SEQLEN + l;
        float dv = delta[row * DINNER + d];
        float xv = xc[row * DINNER + d];
        const float* xd = xdbl + row * XDBL_W;
        float accy = 0.f;
#pragma unroll
        for (int n = 0; n < DSTATE; ++n) {
            float dA = __expf(dv * a[n]);
            h[n] = dA * h[n] + (dv * xv) * xd[DTRANK + n];          // B_n
            accy += h[n] * xd[DTRANK + DSTATE + n];                 // C_n
        }
        float zv = xz[row * (2 * DINNER) + DINNER + d];
        y[row * DINNER + d] = (accy + xv * dcoef) * silu_(zv);
    }
}

// ---------------------------------------------------------------------------
extern "C" void kernel_launch(void* const* d_in, const int* in_sizes, int n_in,
                              void* d_out, int out_size, void* d_ws, size_t ws_size,
                              hipStream_t stream) {
    const float* x    = (const float*)d_in[0];   // [B,L,768]
    const float* Win  = (const float*)d_in[1];   // [3072,768]
    const float* cw   = (const float*)d_in[2];   // [1536,1,4]
    const float* cb   = (const float*)d_in[3];   // [1536]
    const float* Wx   = (const float*)d_in[4];   // [80,1536]
    const float* Wdt  = (const float*)d_in[5];   // [1536,48]
    const float* bdt  = (const float*)d_in[6];   // [1536]
    const float* Alog = (const float*)d_in[7];   // [1536,16]
    const float* Dp   = (const float*)d_in[8];   // [1536]
    const float* Wout = (const float*)d_in[9];   // [768,1536]
    const float* lnw  = (const float*)d_in[10];  // [768]
    const float* lnb  = (const float*)d_in[11];  // [768]
    float* out = (float*)d_out;                  // [B,L,768]

    char* ws = (char*)d_ws;
    float* xn    = (float*)ws;  ws += (size_t)NTOK * DMODEL       * sizeof(float);
    float* xz    = (float*)ws;  ws += (size_t)NTOK * 2 * DINNER   * sizeof(float);
    float* xdbl  = (float*)ws;  ws += (size_t)NTOK * XDBL_W       * sizeof(float);
    float* delta = (float*)ws;  ws += (size_t)NTOK * DINNER       * sizeof(float);
    float* xc    = (float*)ws;  ws += (size_t)NTOK * DINNER       * sizeof(float);

    // 1) LayerNorm
    ln_kernel<<<NTOK / 8, 256, 0, stream>>>(x, lnw, lnb, xn);

    // 2) in_proj: xz = xn @ Win^T  [4096,3072]
    //    tiles = (4096/32)*(3072/64) = 128*48 = 6144 -> 768 blocks
    gemm_wmma_32x64<0><<<6144 / 8, 256, 0, stream>>>(
        xn, DMODEL, Win, DMODEL, xz, 2 * DINNER, (2 * DINNER) / 64, DMODEL,
        nullptr, nullptr, 0);

    // 3) depthwise conv + SiLU -> xc
    conv_silu_kernel<<<(NTOK * DINNER) / 256, 256, 0, stream>>>(xz, cw, cb, xc);

    // 4) x_proj: xdbl = xc @ Wx^T  [4096,80]
    //    tiles = (4096/16)*(80/16) = 256*5 = 1280 -> 160 blocks
    gemm_wmma_16x16<<<1280 / 8, 256, 0, stream>>>(
        xc, DINNER, Wx, DINNER, xdbl, XDBL_W, XDBL_W / 16, DINNER);

    // 5) dt_proj + softplus: delta = softplus(dt @ Wdt^T + bdt)  [4096,1536]
    //    dt = first 48 columns of xdbl (lda = 80).
    //    tiles = 128*(1536/64) = 128*24 = 3072 -> 384 blocks
    gemm_wmma_32x64<1><<<3072 / 8, 256, 0, stream>>>(
        xdbl, XDBL_W, Wdt, DTRANK, delta, DINNER, DINNER / 64, DTRANK,
        bdt, nullptr, 0);

    // 6) selective scan (+ D skip + SiLU(z) gate), y written over xc
    scan_kernel<<<(BATCH * DINNER) / 256, 256, 0, stream>>>(
        delta, xdbl, xz, xc, Alog, Dp, xc);

    // 7) out_proj + residual: out = y @ Wout^T + x  [4096,768]
    //    tiles = 128*(768/64) = 128*12 = 1536 -> 192 blocks
    gemm_wmma_32x64<2><<<1536 / 8, 256, 0, stream>>>(
        xc, DINNER, Wout, DINNER, out, DMODEL, DMODEL / 16 * 16 / 64, DINNER,
        nullptr, x, DMODEL);
}